// Model_38611755991438
// MI455X (gfx1250) — compile-verified
//
#include <hip/hip_runtime.h>

typedef __attribute__((ext_vector_type(2))) float v2f;
typedef __attribute__((ext_vector_type(8))) float v8f;

#define GATES   16384
#define NLAYERS 6
#define BLK     1024
#define IN0     1024
#define ROWS    2

// Constant 4x16 projection W16_TO_4 ([c][j]), matching the reference
// (sequential numpy assignments in row 3 resolved).
constexpr float WT[4][16] = {
  {0,0,0,0, 0,0,0,0, 1,1,1,1, 1,1,1,1},
  {0,0,1,1, 0,0,1,1, -1,-1,0,0, -1,-1,0,0},
  {0,0,0,0, 1,1,1,1, -1,-1,-1,-1, 0,0,0,0},
  {0,1,-1,0, -1,0,-2,-1, 1,2,0,1, 0,1,-1,0}
};

// ---------------------------------------------------------------------------
// Precompute kernel: per layer, w4[c,g] = W16_TO_4 @ softmax(w[:,g]).
// One wave per 16-gate tile; D(16x16) = A(16xK) x B(Kx16) chained over four
// K=4 chunks of V_WMMA_F32_16X16X4_F32 (exact f32).  Straight-line named
// scalars everywhere: no dynamic extractelement, no LDS table.
//   A: lane<16 -> A[M=lane, K={0,1}]; lane>=16 -> K={2,3}
//   B: lane<16 -> B[K={0,1}, N=lane]; lane>=16 -> K={2,3}, N=lane-16
//   D: vgpr r  -> D[M=r (+8 upper half), N=lane&15]
// ---------------------------------------------------------------------------

// 3-cndmask select of W16_TO_4[n][k] over n in 0..3 (k is a literal).
#define BS(k) ((n & 2) ? ((n & 1) ? WT[3][(k)] : WT[2][(k)]) \
                       : ((n & 1) ? WT[1][(k)] : WT[0][(k)]))

#define CHUNK(Q, EA, EB, EC, ED)                                          \
  {                                                                       \
    v2f a, b;                                                             \
    a.x = half ? (EC) : (EA);                                             \
    a.y = half ? (ED) : (EB);                                             \
    b.x = bn4 ? (half ? BS(4 * Q + 2) : BS(4 * Q + 0)) : 0.f;             \
    b.y = bn4 ? (half ? BS(4 * Q + 3) : BS(4 * Q + 1)) : 0.f;             \
    acc = __builtin_amdgcn_wmma_f32_16x16x4_f32(                          \
        false, a, false, b, (short)0, acc, false, false);                 \
  }

__global__ __launch_bounds__(128)
void lgn_coeff_kernel(const float* __restrict__ w, float* __restrict__ coeff)
{
  const int tid  = threadIdx.x;
  const int lane = tid & 31;
  const int wid  = blockIdx.x * 4 + (tid >> 5);   // tile id, 0..1023
  const int g0   = wid << 4;
  const int half = lane >> 4;
  const int n    = lane & 15;
  const bool bn4 = (n < 4);
  const float* wg = w + (g0 + n);

  // 16 logits as named scalars (straight-line).
#define LG(i) float e##i = wg[(i) * GATES];
  LG(0) LG(1) LG(2) LG(3) LG(4) LG(5) LG(6) LG(7)
  LG(8) LG(9) LG(10) LG(11) LG(12) LG(13) LG(14) LG(15)
#undef LG

  float mx = fmaxf(fmaxf(fmaxf(fmaxf(e0, e1), fmaxf(e2, e3)),
                         fmaxf(fmaxf(e4, e5), fmaxf(e6, e7))),
                   fmaxf(fmaxf(fmaxf(e8, e9), fmaxf(e10, e11)),
                         fmaxf(fmaxf(e12, e13), fmaxf(e14, e15))));

#define EX(i) e##i = __expf(e##i - mx);
  EX(0) EX(1) EX(2) EX(3) EX(4) EX(5) EX(6) EX(7)
  EX(8) EX(9) EX(10) EX(11) EX(12) EX(13) EX(14) EX(15)
#undef EX

  const float sum = ((e0 + e1) + (e2 + e3)) + ((e4 + e5) + (e6 + e7)) +
                    ((e8 + e9) + (e10 + e11)) + ((e12 + e13) + (e14 + e15));
  const float inv = 1.f / sum;

#define NM(i) e##i *= inv;
  NM(0) NM(1) NM(2) NM(3) NM(4) NM(5) NM(6) NM(7)
  NM(8) NM(9) NM(10) NM(11) NM(12) NM(13) NM(14) NM(15)
#undef NM

  v8f acc = {0.f, 0.f, 0.f, 0.f, 0.f, 0.f, 0.f, 0.f};
  CHUNK(0, e0,  e1,  e2,  e3)
  CHUNK(1, e4,  e5,  e6,  e7)
  CHUNK(2, e8,  e9,  e10, e11)
  CHUNK(3, e12, e13, e14, e15)

  // D[M=gate_local, N=c]; only N<4 columns are meaningful.
  if (bn4) {
#pragma unroll
    for (int r = 0; r < 8; ++r)
      coeff[(g0 + r + 8 * half) * 4 + n] = acc[r];
  }
}

// ---------------------------------------------------------------------------
// Main kernel: 2 batch rows per workgroup, activations double-buffered in
// 256 KB LDS (4 x 64 KB).  idx/coeff loaded once per gate and applied to both
// rows (halves L2 traffic); gates gather via ds_load; one barrier per layer.
// ---------------------------------------------------------------------------
__global__ __launch_bounds__(BLK)
void lgn_main_kernel(const float* __restrict__ x,
                     const int* __restrict__ i0, const int* __restrict__ i1,
                     const int* __restrict__ i2, const int* __restrict__ i3,
                     const int* __restrict__ i4, const int* __restrict__ i5,
                     const float* __restrict__ coeff,
                     const float* __restrict__ lin_w,
                     const float* __restrict__ lin_b,
                     float* __restrict__ out)
{
  __shared__ float buf[2][ROWS][GATES];   // 256 KB (<= 320 KB per WGP workgroup)
  __shared__ float red[ROWS][BLK / 32];

  const int tid  = threadIdx.x;
  const int row0 = blockIdx.x * ROWS;

  // Layer-0 inputs (IN0 == BLK); layer-0 indices are all < 1024.
  buf[0][0][tid] = x[(size_t)(row0 + 0) * IN0 + tid];
  buf[0][1][tid] = x[(size_t)(row0 + 1) * IN0 + tid];
  __syncthreads();

  const int* idxs[NLAYERS] = {i0, i1, i2, i3, i4, i5};
#pragma unroll
  for (int l = 0; l < NLAYERS; ++l) {
    const int     cur = l & 1;
    const int2*   ip  = (const int2*)idxs[l];
    const float4* cp  = (const float4*)(coeff + (size_t)l * GATES * 4);
    const float*  s0  = buf[cur][0];
    const float*  s1  = buf[cur][1];
    float*        d0  = buf[cur ^ 1][0];
    float*        d1  = buf[cur ^ 1][1];
#pragma unroll 4
    for (int u = 0; u < GATES / BLK; ++u) {
      const int    t = tid + u * BLK;
      const int2   p = ip[t];          // global_load_b64 (coalesced, shared by rows)
      const float4 c = cp[t];          // global_load_b128 (coalesced, shared by rows)
      const float  A0 = s0[p.x];       // ds_load_b32 gathers
      const float  B0 = s0[p.y];
      const float  A1 = s1[p.x];
      const float  B1 = s1[p.y];
      // c0 + c1*A + c2*B + c3*A*B == fma(fma(c3,A,c2), B, fma(c1,A,c0))
      d0[t] = fmaf(fmaf(c.w, A0, c.z), B0, fmaf(c.y, A0, c.x));
      d1[t] = fmaf(fmaf(c.w, A1, c.z), B1, fmaf(c.y, A1, c.x));
    }
    __syncthreads();                   // reads (cur) and writes (cur^1) disjoint
  }

  // After 6 layers the result is back in buf[0].  out[row] = h . lin_w + b.
  const float* h0 = buf[0][0];
  const float* h1 = buf[0][1];
  float p0 = 0.f, p1 = 0.f;
#pragma unroll 4
  for (int t = tid; t < GATES; t += BLK) {
    const float lw = lin_w[t];
    p0 = fmaf(h0[t], lw, p0);
    p1 = fmaf(h1[t], lw, p1);
  }
#pragma unroll
  for (int off = 16; off > 0; off >>= 1) {
    p0 += __shfl_xor(p0, off, 32);
    p1 += __shfl_xor(p1, off, 32);
  }
  if ((tid & 31) == 0) {
    red[0][tid >> 5] = p0;
    red[1][tid >> 5] = p1;
  }
  __syncthreads();
  if (tid < 32) {                      // 32 wave partials, deterministic order
    float v0 = red[0][tid];
    float v1 = red[1][tid];
#pragma unroll
    for (int off = 16; off > 0; off >>= 1) {
      v0 += __shfl_xor(v0, off, 32);
      v1 += __shfl_xor(v1, off, 32);
    }
    if (tid == 0) {
      const float b = lin_b[0];
      out[row0 + 0] = v0 + b;
      out[row0 + 1] = v1 + b;
    }
  }
}

// ---------------------------------------------------------------------------
// Input dict order: x, idx0,w0, idx1,w1, ..., idx5,w5, lin_w, lin_b
// ---------------------------------------------------------------------------
extern "C" void kernel_launch(void* const* d_in, const int* in_sizes, int n_in,
                              void* d_out, int out_size, void* d_ws, size_t ws_size,
                              hipStream_t stream)
{
  const float* x = (const float*)d_in[0];
  const int*   idx[NLAYERS];
  const float* w[NLAYERS];
  for (int l = 0; l < NLAYERS; ++l) {
    idx[l] = (const int*)d_in[1 + 2 * l];
    w[l]   = (const float*)d_in[2 + 2 * l];
  }
  const float* lin_w = (const float*)d_in[13];
  const float* lin_b = (const float*)d_in[14];
  float* out   = (float*)d_out;
  float* coeff = (float*)d_ws;                 // 6 * 16384 * 4 floats = 1.5 MB

  const int B = in_sizes[0] / IN0;             // 2048

  // Batch-independent gate-coefficient precompute (v_wmma_f32_16x16x4_f32 x4).
  for (int l = 0; l < NLAYERS; ++l)
    lgn_coeff_kernel<<<GATES / 16 / 4, 128, 0, stream>>>(
        w[l], coeff + (size_t)l * GATES * 4);

  // LDS-resident gather network, 2 rows per workgroup + final reduction.
  lgn_main_kernel<<<B / ROWS, BLK, 0, stream>>>(
      x, idx[0], idx[1], idx[2], idx[3], idx[4], idx[5],
      coeff, lin_w, lin_b, out);
}